// RelationAttentiveAggregator_532575945235
// MI455X (gfx1250) — compile-verified
//
#include <hip/hip_runtime.h>
#include <math.h>

typedef float v2f __attribute__((ext_vector_type(2)));
typedef float v4f __attribute__((ext_vector_type(4)));
typedef float v8f __attribute__((ext_vector_type(8)));

#define NREL 64
#define DDIM 256
#define RDIM 128
#define TILE_M 64          // edges per block (4 sub-tiles of 16)
#define ASTRIDE 260        // padded row stride (floats); 260*4B is 16B-aligned

// ---------------- workspace layout (bytes) ----------------
// 0      : counts      [64] int
// 256    : cursor      [64] int
// 512    : offsets     [65] int
// 1024   : tileOff     [65] int
// 1536   : stats       [2]  float  (gmax, gsum)
// 2048   : sortedIdx   [E]  int
// 2048+4E: hv          [64*128] float
// then   : scores      [E]  float
// then   : partial     [(E/256)*256] float

__global__ void k_init(int* counts) {
    if (threadIdx.x < NREL) counts[threadIdx.x] = 0;
}

__global__ void k_hist(const int* __restrict__ rel, int* counts, int E) {
    int e = blockIdx.x * blockDim.x + threadIdx.x;
    if (e < E) atomicAdd(&counts[rel[e] & (NREL - 1)], 1);
}

__global__ void k_scan(const int* counts, int* offsets, int* tileOff, int* cursor) {
    if (threadIdx.x == 0) {
        int acc = 0, tacc = 0;
        for (int i = 0; i < NREL; ++i) {
            offsets[i] = acc;
            cursor[i]  = acc;
            tileOff[i] = tacc;
            acc  += counts[i];
            tacc += (counts[i] + TILE_M - 1) / TILE_M;
        }
        offsets[NREL] = acc;
        tileOff[NREL] = tacc;
    }
}

__global__ void k_scatter(const int* __restrict__ rel, int* cursor,
                          int* sortedIdx, int E) {
    int e = blockIdx.x * blockDim.x + threadIdx.x;
    if (e < E) {
        int pos = atomicAdd(&cursor[rel[e] & (NREL - 1)], 1);
        sortedIdx[pos] = e;
    }
}

// hv[n][r] = sum_d head[d] * RM[n][d][r]
__global__ void k_headvals(const float* __restrict__ head,
                           const float* __restrict__ RM,
                           float* __restrict__ hv) {
    int n = blockIdx.x;
    int r = threadIdx.x;                 // 128 threads
    const float* W = RM + (size_t)n * DDIM * RDIM + r;
    float s = 0.f;
#pragma unroll 4
    for (int d = 0; d < DDIM; ++d) s += head[d] * W[d * RDIM];
    hv[n * RDIM + r] = s;
}

// One block = one tile of 64 edges (same relation), 8 waves (256 threads).
// Wave w owns output columns [16w,16w+16); it keeps 4 accumulators (one per
// 16-edge sub-tile) so each B fetch feeds 4 V_WMMA_F32_16X16X4_F32 ops.
__global__ void __launch_bounds__(256)
k_gemm_score(const float* __restrict__ tails,
             const float* __restrict__ RM,
             const float* __restrict__ relEmb,
             const int*  __restrict__ sortedIdx,
             const int*  __restrict__ offsets,
             const int*  __restrict__ tileOff,
             const float* __restrict__ hv,
             float* __restrict__ scores) {
    __shared__ int   sh_n, sh_tile;
    __shared__ int   edges[TILE_M];
    __shared__ __align__(16) float Atile[TILE_M * ASTRIDE];   // 66.5 KB
    __shared__ float part[8][TILE_M];

    int t = threadIdx.x;
    if (t == 0) {
        int b = blockIdx.x;
        int n = -1, tile = 0;
        if (b < tileOff[NREL]) {
            int lo = 0, hi = NREL;       // tileOff[lo] <= b < tileOff[hi]
            while (hi - lo > 1) {
                int mid = (lo + hi) >> 1;
                if (tileOff[mid] <= b) lo = mid; else hi = mid;
            }
            n = lo;
            tile = b - tileOff[lo];
        }
        sh_n = n; sh_tile = tile;
    }
    __syncthreads();
    int n = sh_n;
    if (n < 0) return;

    int start = offsets[n] + sh_tile * TILE_M;
    int end   = offsets[n + 1];
    if (t < TILE_M) edges[t] = (start + t < end) ? sortedIdx[start + t] : -1;
    __syncthreads();

    // stage A (64 edges x 256 dims) into LDS with float4 traffic
    for (int i = t; i < TILE_M * (DDIM / 4); i += 256) {
        int row = i >> 6;                 // /(DDIM/4)
        int c4  = i & 63;
        int e   = edges[row];
        v4f val = {};
        if (e >= 0) val = *(const v4f*)(tails + (size_t)e * DDIM + c4 * 4);
        *(v4f*)(Atile + row * ASTRIDE + c4 * 4) = val;
    }
    __syncthreads();

    int wave = t >> 5;
    int lane = t & 31;
    int nn   = lane & 15;      // N index within 16-col slice (and A's M index)
    int hi   = lane >> 4;      // half-wave select (K split)
    int col0 = wave * 16;

    const float* Bp = RM + (size_t)n * DDIM * RDIM + col0 + nn;   // B[k][col]
    __builtin_prefetch(Bp, 0, 1);
    const float* A0 = Atile + (nn +  0) * ASTRIDE + 2 * hi;
    const float* A1 = Atile + (nn + 16) * ASTRIDE + 2 * hi;
    const float* A2 = Atile + (nn + 32) * ASTRIDE + 2 * hi;
    const float* A3 = Atile + (nn + 48) * ASTRIDE + 2 * hi;

    v8f acc0 = {}, acc1 = {}, acc2 = {}, acc3 = {};
#pragma unroll 4
    for (int k0 = 0; k0 < DDIM; k0 += 4) {
        v2f b;
        b.x = Bp[(k0 + 2 * hi)     * RDIM];
        b.y = Bp[(k0 + 2 * hi + 1) * RDIM];
        v2f a0 = *(const v2f*)(A0 + k0);
        v2f a1 = *(const v2f*)(A1 + k0);
        v2f a2 = *(const v2f*)(A2 + k0);
        v2f a3 = *(const v2f*)(A3 + k0);
        acc0 = __builtin_amdgcn_wmma_f32_16x16x4_f32(false, a0, false, b, (short)0, acc0, false, false);
        acc1 = __builtin_amdgcn_wmma_f32_16x16x4_f32(false, a1, false, b, (short)0, acc1, false, false);
        acc2 = __builtin_amdgcn_wmma_f32_16x16x4_f32(false, a2, false, b, (short)0, acc2, false, false);
        acc3 = __builtin_amdgcn_wmma_f32_16x16x4_f32(false, a3, false, b, (short)0, acc3, false, false);
    }

    // epilogue: tv = tanh(acc + rel_emb), contribution = hv * tv,
    // reduce over the wave's 16 columns with a width-16 shuffle tree.
    float hvv = hv[n * RDIM + col0 + nn];
    float rev = relEmb[n * RDIM + col0 + nn];
    v8f accs[4] = {acc0, acc1, acc2, acc3};
#pragma unroll
    for (int s = 0; s < 4; ++s) {
#pragma unroll
        for (int j = 0; j < 8; ++j) {
            // C/D layout: VGPR j, lanes0-15 -> M=j, lanes16-31 -> M=j+8
            float c = hvv * tanhf(accs[s][j] + rev);
            c += __shfl_xor(c, 8, 16);
            c += __shfl_xor(c, 4, 16);
            c += __shfl_xor(c, 2, 16);
            c += __shfl_xor(c, 1, 16);
            if (nn == 0) part[wave][s * 16 + j + 8 * hi] = c;
        }
    }
    __syncthreads();

    if (t < TILE_M) {
        float s = 0.f;
#pragma unroll
        for (int w = 0; w < 8; ++w) s += part[w][t];
        int e = edges[t];
        if (e >= 0) scores[e] = s;
    }
}

// single-block deterministic softmax stats (max + sum of exp)
__global__ void __launch_bounds__(1024)
k_softmax_stats(const float* __restrict__ scores, float* stats, int E) {
    __shared__ float red[1024];
    int t = threadIdx.x;
    float m = -3.402823466e38f;
    for (int i = t; i < E; i += 1024) m = fmaxf(m, scores[i]);
    red[t] = m; __syncthreads();
    for (int s = 512; s > 0; s >>= 1) {
        if (t < s) red[t] = fmaxf(red[t], red[t + s]);
        __syncthreads();
    }
    float gmax = red[0]; __syncthreads();
    float sum = 0.f;
    for (int i = t; i < E; i += 1024) sum += expf(scores[i] - gmax);
    red[t] = sum; __syncthreads();
    for (int s = 512; s > 0; s >>= 1) {
        if (t < s) red[t] += red[t + s];
        __syncthreads();
    }
    if (t == 0) { stats[0] = gmax; stats[1] = red[0]; }
}

// block b: partial[b][d] = sum over its 256 edges of prob(e)*tail[e][d]
__global__ void __launch_bounds__(256)
k_wsum(const float* __restrict__ tails, const float* __restrict__ scores,
       const float* __restrict__ stats, float* __restrict__ partial) {
    __shared__ float p[256];
    int t = threadIdx.x;
    int base = blockIdx.x * 256;
    float gmax = stats[0];
    float inv  = 1.0f / stats[1];
    p[t] = expf(scores[base + t] - gmax) * inv;
    __syncthreads();
    float acc = 0.f;
    const float* tp = tails + (size_t)base * DDIM + t;   // thread t owns dim d=t
#pragma unroll 4
    for (int i = 0; i < 256; ++i) acc += p[i] * tp[(size_t)i * DDIM];
    partial[blockIdx.x * DDIM + t] = acc;
}

__global__ void k_final(const float* __restrict__ partial, float* __restrict__ out,
                        int nblk) {
    int t = threadIdx.x;                  // 256 threads, one per dim
    float s = 0.f;
    for (int b = 0; b < nblk; ++b) s += partial[b * DDIM + t];
    out[t] = s;
}

extern "C" void kernel_launch(void* const* d_in, const int* in_sizes, int n_in,
                              void* d_out, int out_size, void* d_ws, size_t ws_size,
                              hipStream_t stream) {
    const float* head   = (const float*)d_in[0];
    const float* tails  = (const float*)d_in[1];
    const int*   rel    = (const int*)d_in[2];
    const float* RM     = (const float*)d_in[3];
    const float* relEmb = (const float*)d_in[4];
    float* out = (float*)d_out;

    int E = in_sizes[2];                  // 16384

    char* ws = (char*)d_ws;
    int*   counts    = (int*)(ws + 0);
    int*   cursor    = (int*)(ws + 256);
    int*   offsets   = (int*)(ws + 512);
    int*   tileOff   = (int*)(ws + 1024);
    float* stats     = (float*)(ws + 1536);
    int*   sortedIdx = (int*)(ws + 2048);
    float* hv        = (float*)(ws + 2048 + (size_t)E * 4);
    float* scores    = hv + NREL * RDIM;
    float* partial   = scores + E;

    int nEdgeBlk = (E + 255) / 256;
    int maxTiles = (E + TILE_M - 1) / TILE_M + NREL;

    k_init<<<1, 64, 0, stream>>>(counts);
    k_hist<<<nEdgeBlk, 256, 0, stream>>>(rel, counts, E);
    k_scan<<<1, 32, 0, stream>>>(counts, offsets, tileOff, cursor);
    k_scatter<<<nEdgeBlk, 256, 0, stream>>>(rel, cursor, sortedIdx, E);
    k_headvals<<<NREL, RDIM, 0, stream>>>(head, RM, hv);
    k_gemm_score<<<maxTiles, 256, 0, stream>>>(tails, RM, relEmb, sortedIdx,
                                               offsets, tileOff, hv, scores);
    k_softmax_stats<<<1, 1024, 0, stream>>>(scores, stats, E);
    k_wsum<<<nEdgeBlk, 256, 0, stream>>>(tails, scores, stats, partial);
    k_final<<<1, DDIM, 0, stream>>>(partial, out, nEdgeBlk);
}